// GATLayer_8641474200164
// MI455X (gfx1250) — compile-verified
//
#include <hip/hip_runtime.h>

typedef __attribute__((ext_vector_type(2))) float v2f;
typedef __attribute__((ext_vector_type(8))) float v8f;

#define B_ 8
#define N_ 2048
#define F_ 128
#define ALPHA 0.2f
#define NEG_BIG (-9000000000000000.0f)

// LDS h-chunk layout: [col][j] with padded j-stride (even => 8B-aligned v2f
// reads; 34 avoids systematic 32-bank stride conflicts).
#define JCH 32
#define JSTR 34

// ---------------------------------------------------------------------------
// Kernel 1: h[b,n,o] = sum_f input[b,n,f] * W[o,f]
// grid = B*N/16 blocks, 128 threads (4 waves). Wave w owns output cols
// [w*32, w*32+32) as two 16x16 WMMA tiles; K-loop over F in steps of 4 using
// V_WMMA_F32_16X16X4_F32 (exact f32 matrix math, matches reference dtype).
//
// f32 A 16x4 layout (ISA 7.12.2): lanes 0-15 hold M=lane with VGPR0/1 = K0/K1,
// lanes 16-31 hold M=lane-16 with VGPR0/1 = K2/K3.  B 4x16 mirrors it with
// N striped across lanes.  C/D: VGPR r = row r (lanes 0-15) / row 8+r.
// ---------------------------------------------------------------------------
__global__ __launch_bounds__(128) void gemm_h_kernel(
    const float* __restrict__ inp, const float* __restrict__ W,
    float* __restrict__ h) {
  const int tile = blockIdx.x;
  const int b    = tile / (N_ / 16);
  const int r0   = (tile % (N_ / 16)) * 16;
  const int lane = threadIdx.x & 31;
  const int w    = threadIdx.x >> 5;
  const int half = lane >> 4;
  const int lrow = lane & 15;
  const int co   = w * 32;

  const float* Arow = inp + ((size_t)b * N_ + r0 + lrow) * F_;
  const float* W0   = W + (size_t)(co + lrow) * F_;
  const float* W1   = W + (size_t)(co + 16 + lrow) * F_;

  v8f acc0 = {}; v8f acc1 = {};
  for (int k0 = 0; k0 < F_; k0 += 4) {
    const int ka = k0 + 2 * half;
    v2f afrag = *(const v2f*)(Arow + ka);
    v2f b0    = *(const v2f*)(W0 + ka);
    v2f b1    = *(const v2f*)(W1 + ka);
    acc0 = __builtin_amdgcn_wmma_f32_16x16x4_f32(false, afrag, false, b0,
                                                 (short)0, acc0, false, false);
    acc1 = __builtin_amdgcn_wmma_f32_16x16x4_f32(false, afrag, false, b1,
                                                 (short)0, acc1, false, false);
  }

  float* hb = h + ((size_t)b * N_ + r0) * F_;
#pragma unroll
  for (int r = 0; r < 8; ++r) {
    const int m = r + 8 * half;
    hb[(size_t)m * F_ + co + lrow]      = acc0[r];
    hb[(size_t)m * F_ + co + 16 + lrow] = acc1[r];
  }
}

// ---------------------------------------------------------------------------
// Kernel 2: s1[row] = h[row,:].a[:128], s2[row] = h[row,:].a[128:256]
// ---------------------------------------------------------------------------
__global__ __launch_bounds__(256) void s_kernel(
    const float* __restrict__ h, const float* __restrict__ a,
    float* __restrict__ s1, float* __restrict__ s2) {
  const int idx = blockIdx.x * blockDim.x + threadIdx.x;
  if (idx >= B_ * N_) return;
  const float* hr = h + (size_t)idx * F_;
  float acc1 = 0.0f, acc2 = 0.0f;
#pragma unroll 4
  for (int f = 0; f < F_; ++f) {
    const float v = hr[f];
    acc1 = __builtin_fmaf(v, a[f], acc1);
    acc2 = __builtin_fmaf(v, a[F_ + f], acc2);
  }
  s1[idx] = acc1;
  s2[idx] = acc2;
}

// ---------------------------------------------------------------------------
// Kernel 3: fused masked-softmax + (att @ h).  One block = 32 rows of one
// batch, 256 threads (8 waves).
//  Pass 1: per-row online max/sum-exp over all 2048 masked scores.
//  Pass 2: j-chunks of 32: the h chunk (32x128) is staged once per block into
//  LDS (coalesced global reads, stored [col][j]-major so every WMMA
//  B-fragment is one aligned ds_load_b64), while the next chunk is prefetched
//  (global_prefetch_b8).  Probabilities are recomputed in-register as WMMA
//  A-fragments (p = exp(val-M)/S; masked -> NEG_BIG -> exp underflows to 0;
//  the all-masked-row limit stays exact) and accumulated with f32 WMMA.
//  The (B,N,N) attention tensor never exists in memory.
// ---------------------------------------------------------------------------
__global__ __launch_bounds__(256) void attn_kernel(
    const float* __restrict__ h, const int* __restrict__ adj,
    const float* __restrict__ s1, const float* __restrict__ s2,
    float* __restrict__ out) {
  __shared__ float sh_h[F_ * JSTR];   // staged h chunk, [col][j] padded
  __shared__ float pm[32][8];
  __shared__ float ps[32][8];
  __shared__ float Mrow[32];
  __shared__ float iSrow[32];

  const int blk   = blockIdx.x;
  const int b     = blk / (N_ / 32);
  const int rbase = (blk % (N_ / 32)) * 32;
  const int t     = threadIdx.x;

  const float* s1b  = s1 + (size_t)b * N_;
  const float* s2b  = s2 + (size_t)b * N_;
  const int*   adjb = adj + (size_t)b * N_ * N_;
  const float* hb   = h + (size_t)b * N_ * F_;

  // ---- Pass 1: masked row max + sum(exp) -------------------------------
  {
    const int lr  = t >> 3;   // 0..31 local row
    const int q   = t & 7;    // 8 threads cooperate on one row
    const int row = rbase + lr;
    const float s1r  = s1b[row];
    const int* arow  = adjb + (size_t)row * N_;
    float m = -__builtin_inff();
    float s = 0.0f;
    for (int j = q * 4; j < N_; j += 32) {
      const int4   av  = *(const int4*)(arow + j);
      const float4 s2v = *(const float4*)(s2b + j);
      const int   ai[4] = {av.x, av.y, av.z, av.w};
      const float sv[4] = {s2v.x, s2v.y, s2v.z, s2v.w};
#pragma unroll
      for (int u = 0; u < 4; ++u) {
        float e = s1r + sv[u];
        e = e > 0.0f ? e : ALPHA * e;
        const float val = ai[u] > 0 ? e : NEG_BIG;
        if (val > m) {
          s = __builtin_fmaf(s, __expf(m - val), 1.0f);
          m = val;
        } else {
          s += __expf(val - m);
        }
      }
    }
    pm[lr][q] = m;
    ps[lr][q] = s;
  }
  __syncthreads();
  if (t < 32) {
    float M = -__builtin_inff();
#pragma unroll
    for (int q = 0; q < 8; ++q) M = fmaxf(M, pm[t][q]);
    float S = 0.0f;
#pragma unroll
    for (int q = 0; q < 8; ++q) S = __builtin_fmaf(ps[t][q], __expf(pm[t][q] - M), S);
    Mrow[t]  = M;
    iSrow[t] = 1.0f / S;
  }
  __syncthreads();

  // ---- Pass 2: out[rows, cols] += p * h --------------------------------
  const int lane = t & 31;
  const int w    = t >> 5;
  const int half = lane >> 4;
  const int lrow = lane & 15;
  const int rloc = 16 * (w >> 2) + lrow;   // local row carried by this lane
  const int row  = rbase + rloc;
  const int co   = (w & 3) * 32;

  const float M   = Mrow[rloc];
  const float iS  = iSrow[rloc];
  const float s1r = s1b[row];
  const int* arow = adjb + (size_t)row * N_;

  v8f acc0 = {}; v8f acc1 = {};

  for (int jc = 0; jc < N_; jc += JCH) {
    // Stage h[jc..jc+31][0..127] into LDS, transposed to [col][j]-major.
    // Global side: fully coalesced (consecutive threads read consecutive
    // floats).  Also prefetch the next chunk into the cache hierarchy.
    __syncthreads();
    {
      const float* src = hb + (size_t)jc * F_;
#pragma unroll
      for (int i = 0; i < (JCH * F_) / 256; ++i) {
        const int e   = i * 256 + t;
        const int jj  = e >> 7;      // j within chunk
        const int col = e & (F_ - 1);
        sh_h[col * JSTR + jj] = src[e];
      }
      if (jc + JCH < N_)
        __builtin_prefetch(src + JCH * F_ + t * 16, 0, 1);  // global_prefetch_b8
    }
    __syncthreads();

    for (int jq = 0; jq < JCH; jq += 4) {
      const int ja = jc + jq + 2 * half;
      // A fragment: p values for (row, ja), (row, ja+1)
      v2f afrag;
#pragma unroll
      for (int u = 0; u < 2; ++u) {
        const int j = ja + u;
        float e = s1r + s2b[j];
        e = e > 0.0f ? e : ALPHA * e;
        const float val = arow[j] > 0 ? e : NEG_BIG;
        afrag[u] = __expf(val - M) * iS;   // masked -> exp underflow -> 0
      }
      // B fragments: one aligned 8-byte LDS read per tile per lane.
      const int jb = jq + 2 * half;
      v2f b0 = *(const v2f*)&sh_h[(co + lrow) * JSTR + jb];
      v2f b1 = *(const v2f*)&sh_h[(co + 16 + lrow) * JSTR + jb];
      acc0 = __builtin_amdgcn_wmma_f32_16x16x4_f32(false, afrag, false, b0,
                                                   (short)0, acc0, false, false);
      acc1 = __builtin_amdgcn_wmma_f32_16x16x4_f32(false, afrag, false, b1,
                                                   (short)0, acc1, false, false);
    }
  }

  float* ob = out + ((size_t)b * N_ + rbase + 16 * (w >> 2)) * F_;
#pragma unroll
  for (int r = 0; r < 8; ++r) {
    const int m2 = r + 8 * half;
    ob[(size_t)m2 * F_ + co + lrow]      = acc0[r];
    ob[(size_t)m2 * F_ + co + 16 + lrow] = acc1[r];
  }
}

// ---------------------------------------------------------------------------
extern "C" void kernel_launch(void* const* d_in, const int* in_sizes, int n_in,
                              void* d_out, int out_size, void* d_ws,
                              size_t ws_size, hipStream_t stream) {
  const float* inp = (const float*)d_in[0];   // (B, N, F_IN)  f32
  const int*   adj = (const int*)d_in[1];     // (B, N, N)     i32
  const float* W   = (const float*)d_in[2];   // (F_OUT, F_IN) f32
  const float* a   = (const float*)d_in[3];   // (1, 2*F_OUT)  f32
  float* out = (float*)d_out;                 // (B, N, F_OUT) f32

  float* h  = (float*)d_ws;                   // B*N*F floats (8 MB)
  float* s1 = h + (size_t)B_ * N_ * F_;       // B*N floats
  float* s2 = s1 + (size_t)B_ * N_;           // B*N floats

  gemm_h_kernel<<<B_ * N_ / 16, 128, 0, stream>>>(inp, W, h);
  s_kernel<<<(B_ * N_ + 255) / 256, 256, 0, stream>>>(h, a, s1, s2);
  attn_kernel<<<B_ * N_ / 32, 256, 0, stream>>>(h, adj, s1, s2, out);
}